// SVHPerm_21629455302820
// MI455X (gfx1250) — compile-verified
//
#include <hip/hip_runtime.h>
#include <hip/hip_bf16.h>
#include <math.h>

typedef __attribute__((ext_vector_type(2))) float v2f;
typedef __attribute__((ext_vector_type(8))) float v8f;

#define NN    40
#define AB    1600      // 40*40
#define N4    2560000   // 40^4
#define WID   128
#define KC    64        // K-chunk staged in LDS per block
#define LDSP  (KC + 4)  // row stride 68 -> bank stride 4 mod 64, conflict-free

// ---------------------------------------------------------------------------
// fp32 WMMA GEMM: C[M,N] = op(A)[M,K] @ B[K,N] (+ bias_basis bias, leaky)
//   FOLD=false: A[m*lda + k]
//   FOLD=true : A[(k>>7)*ablkStride + m*lda + (k&127)]   (t[i,k,ab] gather)
// Block = wpb waves; each wave owns a 16x16 tile of one 16-row stripe.
// The 16 x KC A-tile is staged through LDS (shared by all waves of the block);
// B is read directly (16-lane contiguous segments) with pointer stepping.
// V_WMMA_F32_16X16X4_F32, 16 MMAs per chunk, EXEC all-1s throughout.
// ---------------------------------------------------------------------------
template <bool FOLD>
__global__ __launch_bounds__(256)
void gemm_wmma_f32(const float* __restrict__ A,
                   const float* __restrict__ B,
                   float* __restrict__ C,
                   int M, int N, int K,
                   int lda, int ldb, int ldc,
                   long long ablkStride,
                   const float* __restrict__ bb,
                   const float* __restrict__ bw,
                   int leaky)
{
    __shared__ float As[16 * LDSP];

    const int tid  = threadIdx.x;
    const int lane = tid & 31;
    const int wave = tid >> 5;
    const int wpb  = blockDim.x >> 5;
    const int m0 = blockIdx.x * 16;
    const int n0 = (blockIdx.y * wpb + wave) * 16;

    const int mrow  = lane & 15;         // A frag row
    const int ncol  = lane & 15;         // B/C frag col
    const int khalf = (lane >> 4) << 1;  // lanes 0-15: K 0,1 ; lanes 16-31: K 2,3

    v8f acc = {};
    for (int kbase = 0; kbase < K; kbase += KC) {
        const int kcnt = (K - kbase < KC) ? (K - kbase) : KC;

        __syncthreads();  // previous chunk fully consumed before overwrite
        // --- stage A[m0..m0+15, kbase..kbase+kcnt) into LDS, coalesced ---
        for (int e = tid; e < 16 * KC; e += blockDim.x) {
            const int r = e >> 6;        // KC == 64
            const int c = e & 63;
            float v = 0.f;
            if (c < kcnt) {
                long long off;
                if (FOLD) {
                    const int kg = kbase + c;          // chunk never crosses a
                    off = (long long)(kg >> 7) * ablkStride +  // 128-block (kbase%64==0)
                          (long long)(m0 + r) * lda + (kg & 127);
                } else {
                    off = (long long)(m0 + r) * lda + (kbase + c);
                }
                v = A[off];
            }
            As[r * LDSP + c] = v;
        }
        // hint-prefetch the next A chunk (lowers to global_prefetch_b8)
        if (kbase + KC < K) {
            const int r = (tid >> 6) & 15;
            const int c = (tid & 63);
            long long off;
            const int kg = kbase + KC + c;
            if (FOLD)
                off = (long long)(kg >> 7) * ablkStride +
                      (long long)(m0 + r) * lda + (kg & 127);
            else
                off = (long long)(m0 + r) * lda + kg;
            __builtin_prefetch(&A[off], 0, 1);
        }
        __syncthreads();

        // --- MMA over the staged chunk ---
        const int steps = kcnt >> 2;
        const float* bptr = B + (long long)(kbase + khalf) * ldb + n0 + ncol;
        const float* ap   = &As[mrow * LDSP + khalf];
#pragma unroll 4
        for (int s = 0; s < steps; ++s) {
            v2f afrag; afrag.x = ap[0]; afrag.y = ap[1];
            v2f bfrag; bfrag.x = bptr[0]; bfrag.y = bptr[ldb];
            ap   += 4;
            bptr += 4 * (long long)ldb;
            acc = __builtin_amdgcn_wmma_f32_16x16x4_f32(false, afrag, false, bfrag,
                                                        (short)0, acc, false, false);
        }
    }

    // 16x16 f32 C/D layout: VGPR r -> M = r (+8 for lanes 16-31), N = lane&15
    const int rowAdd = (lane >> 4) << 3;
#pragma unroll
    for (int r = 0; r < 8; ++r) {
        const int m = m0 + r + rowAdd;
        const int n = n0 + ncol;
        float v = acc[r];
        if (bw) v += bw[2 * n] * bb[m] + bw[2 * n + 1] * bb[AB + m];
        if (leaky) v = (v > 0.f) ? v : 0.01f * v;
        C[(long long)m * ldc + n] = v;
    }
}

// X0[a*40+b] = sum_d S[a,d]*S[b,d]
__global__ void x0_kernel(const float* __restrict__ S, float* __restrict__ X0)
{
    int idx = blockIdx.x * blockDim.x + threadIdx.x;
    if (idx >= AB) return;
    int a = idx / NN, b = idx % NN;
    float s = 0.f;
    for (int d = 0; d < 5; ++d) s += S[a * 5 + d] * S[b * 5 + d];
    X0[idx] = s;
}

// Layer-0 GEMV: T0[ab*16+k] = dot(Bsym[k,ab,:], X0)  (k >= Ksym -> 0 pad)
__global__ void t0_kernel(const float* __restrict__ Bsym, const float* __restrict__ X0,
                          float* __restrict__ T0, int Ksym)
{
    int wid  = (blockIdx.x * blockDim.x + threadIdx.x) >> 5;
    int lane = threadIdx.x & 31;
    if (wid >= AB * 16) return;
    int ab = wid >> 4;
    int k  = wid & 15;
    float s = 0.f;
    if (k < Ksym) {
        const float* row = Bsym + (long long)k * N4 + (long long)ab * AB;
        for (int cd = lane; cd < AB; cd += 32) s += row[cd] * X0[cd];
    }
    for (int off = 16; off > 0; off >>= 1) s += __shfl_xor(s, off, 32);
    if (lane == 0) T0[ab * 16 + k] = s;
}

// W0mat[k*128+o] = W0[o,0,k] (pad k to 16);  Wpmat[i*16+k] = Wp[0,i,k] (pad)
__global__ void prep_small(const float* __restrict__ W0, const float* __restrict__ Wp,
                           float* __restrict__ W0m, float* __restrict__ Wpm, int Ksym)
{
    int idx = blockIdx.x * blockDim.x + threadIdx.x;
    if (idx < 16 * WID) {
        int k = idx >> 7, o = idx & 127;
        W0m[idx] = (k < Ksym) ? W0[o * Ksym + k] : 0.f;
        int i = idx >> 4, kk = idx & 15;
        Wpm[idx] = (kk < Ksym) ? Wp[i * Ksym + kk] : 0.f;
    }
}

// W2[(k*128+i)*128+o] = Wh[l][o][i][k]
__global__ void prep_w2(const float* __restrict__ Whl, float* __restrict__ W2, int Kfull)
{
    int idx = blockIdx.x * blockDim.x + threadIdx.x;
    if (idx >= Kfull * WID * WID) return;
    int o = idx & 127;
    int i = (idx >> 7) & 127;
    int k = idx >> 14;
    W2[idx] = Whl[(o * WID + i) * Kfull + k];
}

// pairs[ab] = sum_k sum_cd Bsym[k,cd,ab] * YT[cd*16+k] + bp.bias_basis
__global__ void pairs_kernel(const float* __restrict__ Bsym, const float* __restrict__ YT,
                             const float* __restrict__ bb, const float* __restrict__ bp,
                             float* __restrict__ pairs, int Ksym)
{
    int ab  = blockIdx.x;
    int tid = threadIdx.x;
    float s = 0.f;
    for (int k = 0; k < Ksym; ++k) {
        const float* base = Bsym + (long long)k * N4 + ab;
        for (int cd = tid; cd < AB; cd += blockDim.x)
            s += base[(long long)cd * AB] * YT[cd * 16 + k];
    }
    __shared__ float red[256];
    red[tid] = s; __syncthreads();
    for (int off = 128; off > 0; off >>= 1) {
        if (tid < off) red[tid] += red[tid + off];
        __syncthreads();
    }
    if (tid == 0) pairs[ab] = red[0] + bp[0] * bb[ab] + bp[1] * bb[AB + ab];
}

// scalar = bi + sum_{ab,i} X[ab,i]*(Wi[i,0]*bb0[ab] + Wi[i,1]*bb1[ab])
__global__ void identity_kernel(const float* __restrict__ X, const float* __restrict__ Wi,
                                const float* __restrict__ bb, const float* __restrict__ bi,
                                float* __restrict__ outsc)
{
    int tid = threadIdx.x;
    float s = 0.f;
    for (int idx = tid; idx < AB * WID; idx += blockDim.x) {
        int i = idx & 127, ab = idx >> 7;
        s += X[idx] * (Wi[2 * i] * bb[ab] + Wi[2 * i + 1] * bb[AB + ab]);
    }
    __shared__ float red[256];
    red[tid] = s; __syncthreads();
    for (int off = 128; off > 0; off >>= 1) {
        if (tid < off) red[tid] += red[tid + off];
        __syncthreads();
    }
    if (tid == 0) outsc[0] = red[0] + bi[0];
}

// A5[c,d] = sum_{a,b} pairs[a,b]*S[a,c]*S[b,d] + scalar*I
__global__ void a5_kernel(const float* __restrict__ pairs, const float* __restrict__ S,
                          const float* __restrict__ sc, float* __restrict__ A5)
{
    int t = threadIdx.x;
    if (t >= 25) return;
    int c = t / 5, d = t % 5;
    float s = 0.f;
    for (int a = 0; a < NN; ++a) {
        float sac = S[a * 5 + c];
        for (int b = 0; b < NN; ++b) s += pairs[a * NN + b] * sac * S[b * 5 + d];
    }
    if (c == d) s += sc[0];
    A5[t] = s;
}

// Cyclic Jacobi eigh on symmetrized 5x5; out = S @ eigvec(max eigenvalue)
__global__ void eig_out_kernel(const float* __restrict__ A5, const float* __restrict__ S,
                               float* __restrict__ out)
{
    __shared__ float u[5];
    if (threadIdx.x == 0) {
        float Mm[5][5], V[5][5];
        for (int i = 0; i < 5; ++i)
            for (int j = 0; j < 5; ++j) {
                Mm[i][j] = 0.5f * (A5[i * 5 + j] + A5[j * 5 + i]);
                V[i][j] = (i == j) ? 1.f : 0.f;
            }
        for (int sweep = 0; sweep < 40; ++sweep)
            for (int p = 0; p < 4; ++p)
                for (int q = p + 1; q < 5; ++q) {
                    float apq = Mm[p][q];
                    if (fabsf(apq) < 1e-20f) continue;
                    float th = 0.5f * (Mm[q][q] - Mm[p][p]) / apq;
                    float t = 1.f / (fabsf(th) + sqrtf(th * th + 1.f));
                    if (th < 0.f) t = -t;
                    float c = 1.f / sqrtf(t * t + 1.f);
                    float sn = t * c;
                    for (int k = 0; k < 5; ++k) {
                        float a = Mm[k][p], b = Mm[k][q];
                        Mm[k][p] = c * a - sn * b; Mm[k][q] = sn * a + c * b;
                    }
                    for (int k = 0; k < 5; ++k) {
                        float a = Mm[p][k], b = Mm[q][k];
                        Mm[p][k] = c * a - sn * b; Mm[q][k] = sn * a + c * b;
                    }
                    for (int k = 0; k < 5; ++k) {
                        float a = V[k][p], b = V[k][q];
                        V[k][p] = c * a - sn * b; V[k][q] = sn * a + c * b;
                    }
                }
        int best = 0; float bv = Mm[0][0];
        for (int i = 1; i < 5; ++i) if (Mm[i][i] > bv) { bv = Mm[i][i]; best = i; }
        for (int i = 0; i < 5; ++i) u[i] = V[i][best];
    }
    __syncthreads();
    int a = threadIdx.x;
    if (a < NN) {
        float s = 0.f;
        for (int c = 0; c < 5; ++c) s += S[a * 5 + c] * u[c];
        out[a] = s;
    }
}

extern "C" void kernel_launch(void* const* d_in, const int* in_sizes, int n_in,
                              void* d_out, int out_size, void* d_ws, size_t ws_size,
                              hipStream_t stream)
{
    (void)n_in; (void)out_size; (void)ws_size;
    const float* S     = (const float*)d_in[0];
    const float* Bsym  = (const float*)d_in[1];
    const float* Bfull = (const float*)d_in[2];
    const float* bb    = (const float*)d_in[3];
    const float* W0    = (const float*)d_in[4];
    const float* b0    = (const float*)d_in[5];
    const float* Wh    = (const float*)d_in[6];
    const float* bh    = (const float*)d_in[7];
    const float* Wp    = (const float*)d_in[8];
    const float* bp    = (const float*)d_in[9];
    const float* Wi    = (const float*)d_in[10];
    const float* bi    = (const float*)d_in[11];

    const int Ksym  = in_sizes[1] / N4;                    // 11
    const int Kfull = in_sizes[2] / N4;                    // 15
    const int nhid  = in_sizes[6] / (WID * WID * Kfull);   // 2

    float* ws  = (float*)d_ws;
    float* X0  = ws; ws += AB;
    float* T0  = ws; ws += AB * 16;
    float* Xa  = ws; ws += AB * WID;
    float* Xb  = ws; ws += AB * WID;
    float* C1  = ws; ws += (size_t)Kfull * AB * WID;       // t[i,k,ab] as [(k,ab), i]
    float* W2  = ws; ws += (size_t)Kfull * WID * WID;
    float* W0m = ws; ws += 16 * WID;
    float* Wpm = ws; ws += WID * 16;
    float* YT  = ws; ws += AB * 16;
    float* pr  = ws; ws += AB;
    float* sc  = ws; ws += 8;
    float* A5  = ws; ws += 32;

    x0_kernel<<<(AB + 255) / 256, 256, 0, stream>>>(S, X0);
    prep_small<<<(16 * WID + 255) / 256, 256, 0, stream>>>(W0, Wp, W0m, Wpm, Ksym);
    t0_kernel<<<(AB * 16) / 8, 256, 0, stream>>>(Bsym, X0, T0, Ksym);

    // Layer 0: X = leaky(T0(1600x16) @ W0m(16x128) + bias)
    gemm_wmma_f32<false><<<dim3(AB / 16, 1), 256, 0, stream>>>(
        T0, W0m, Xa, AB, WID, 16, 16, WID, WID, 0, bb, b0, 1);

    float* cur = Xa; float* nxt = Xb;
    for (int l = 0; l < nhid; ++l) {
        prep_w2<<<(Kfull * WID * WID + 255) / 256, 256, 0, stream>>>(
            Wh + (size_t)l * WID * WID * Kfull, W2, Kfull);
        // t: Bfull(24000x1600) @ X(1600x128) -> C1(24000x128)
        gemm_wmma_f32<false><<<dim3(Kfull * AB / 16, 1), 256, 0, stream>>>(
            Bfull, cur, C1, Kfull * AB, WID, AB, AB, WID, WID, 0,
            nullptr, nullptr, 0);
        // X' = leaky(fold(C1)(1600x1920) @ W2(1920x128) + bias)
        gemm_wmma_f32<true><<<dim3(AB / 16, 1), 256, 0, stream>>>(
            C1, W2, nxt, AB, WID, Kfull * WID, WID, WID, WID,
            (long long)AB * WID, bb, bh + (size_t)l * WID * 2, 1);
        float* t = cur; cur = nxt; nxt = t;
    }

    // YT(1600x16) = X(1600x128) @ Wpm(128x16)
    gemm_wmma_f32<false><<<dim3(AB / 16, 1), 32, 0, stream>>>(
        cur, Wpm, YT, AB, 16, WID, WID, 16, 16, 0, nullptr, nullptr, 0);

    pairs_kernel<<<AB, 256, 0, stream>>>(Bsym, YT, bb, bp, pr, Ksym);
    identity_kernel<<<1, 256, 0, stream>>>(cur, Wi, bb, bi, sc);
    a5_kernel<<<1, 32, 0, stream>>>(pr, S, sc, A5);
    eig_out_kernel<<<1, 64, 0, stream>>>(A5, S, (float*)d_out);
}